// GNN_51951924412638
// MI455X (gfx1250) — compile-verified
//
#include <hip/hip_runtime.h>

typedef __attribute__((ext_vector_type(2))) float v2f;
typedef __attribute__((ext_vector_type(8))) float v8f;

// CDNA5 no-return float atomic-add: RMW at L2, tracked on STOREcnt.
// Guarantees global_atomic_add_f32 (HIP's atomicAdd may emit a CAS loop).
__device__ __forceinline__ void atomic_add_f32_noret(float* p, float v) {
#if defined(__HIP_DEVICE_COMPILE__)
  asm volatile("global_atomic_add_f32 %0, %1, off scope:SCOPE_DEV"
               :: "v"(p), "v"(v)
               : "memory");
#else
  atomicAdd(p, v);
#endif
}

// ---------------------------------------------------------------- degree pass
__global__ void k_init_deg(float* __restrict__ deg, int n) {
  int i = blockIdx.x * blockDim.x + threadIdx.x;
  if (i < n) deg[i] = 1.0f;                       // self-loop contribution
}

__global__ void k_deg_edges(const int* __restrict__ dst, float* __restrict__ deg, int E) {
  int i = blockIdx.x * blockDim.x + threadIdx.x;
  if (i < E) atomic_add_f32_noret(&deg[dst[i]], 1.0f);
}

__global__ void k_rsqrt_inplace(float* __restrict__ deg, int n) {
  int i = blockIdx.x * blockDim.x + threadIdx.x;
  if (i < n) deg[i] = rsqrtf(deg[i]);             // deg >= 1 always (self-loop)
}

// ------------------------------------------------- layer-1 dense: h1 = x @ W1
// One wave per 16-node tile. V_WMMA_F32_16X16X4_F32, K=3 padded to 4.
// Branchless operand fill; unconditional coalesced stores for full tiles.
__global__ void k_xw1(const float* __restrict__ x, const float* __restrict__ W1,
                      float* __restrict__ h1lin, int n, int ntiles) {
  int wave = (blockIdx.x * blockDim.x + threadIdx.x) >> 5;
  int lane = threadIdx.x & 31;
  if (wave >= ntiles) return;                     // uniform per wave
  int nb  = wave << 4;
  int r   = lane & 15;
  int hi  = lane >> 4;                            // 0: K=0,1   1: K=2,3
  int row = nb + r; if (row >= n) row = n - 1;    // clamped (full tiles anyway)

  // A (16x4 f32): lanes 0-15 hold K0/K1, lanes 16-31 hold K2/K3 (K3 padded 0)
  float xa = x[row * 3 + (hi << 1)];              // K0 or K2
  float xb = x[row * 3 + 1 + hi];                 // K1 or K2 (always in-bounds)
  // B (4x16 f32): same K split per lane group; W1 row 3 padded 0
  float ba = W1[(hi << 1) * 16 + r];              // W1 row 0 or 2
  float bb = W1[(1 + hi) * 16 + r];               // W1 row 1 or 2 (in-bounds)
  v2f a, b;
  a.x = xa; a.y = hi ? 0.0f : xb;
  b.x = ba; b.y = hi ? 0.0f : bb;

  v8f c = {};
  c = __builtin_amdgcn_wmma_f32_16x16x4_f32(false, a, false, b, (short)0, c, false, false);

  float* outp = h1lin + (size_t)(nb + (hi << 3)) * 16 + r;   // row nb+hi*8, col r
  if (nb + 16 <= n) {                              // wave-uniform full tile
#pragma unroll
    for (int v = 0; v < 8; ++v) outp[(size_t)v * 16] = c[v];
  } else {
#pragma unroll
    for (int v = 0; v < 8; ++v) {
      int orow = nb + (hi << 3) + v;
      if (orow < n) h1lin[(size_t)orow * 16 + r] = c[v];
    }
  }
}

// --------------------------------- layer-2 dense: z2 = relu(h1agg) @ W2 (16x8)
// 4 chained WMMA f32 16x16x4 (K=16); B cols 8..15 zero-padded branchlessly.
__global__ void k_h2(const float* __restrict__ h1agg, const float* __restrict__ W2,
                     float* __restrict__ z2, int n, int ntiles) {
  int wave = (blockIdx.x * blockDim.x + threadIdx.x) >> 5;
  int lane = threadIdx.x & 31;
  if (wave >= ntiles) return;
  int nb  = wave << 4;
  int r   = lane & 15;
  int r7  = r & 7;                                 // safe W2 column
  int hi  = lane >> 4;
  int row = nb + r; if (row >= n) row = n - 1;
  bool valid_col = (r < 8);

  v8f c = {};
#pragma unroll
  for (int kc = 0; kc < 4; ++kc) {
    int k0 = (kc << 2) + (hi << 1);
    v2f a, b;
    a.x = fmaxf(h1agg[(size_t)row * 16 + k0 + 0], 0.0f);     // fused ReLU
    a.y = fmaxf(h1agg[(size_t)row * 16 + k0 + 1], 0.0f);
    float w0 = W2[(k0 + 0) * 8 + r7];
    float w1 = W2[(k0 + 1) * 8 + r7];
    b.x = valid_col ? w0 : 0.0f;
    b.y = valid_col ? w1 : 0.0f;
    c = __builtin_amdgcn_wmma_f32_16x16x4_f32(false, a, false, b, (short)0, c, false, false);
  }

  if (valid_col) {
    float* outp = z2 + (size_t)(nb + (hi << 3)) * 8 + r;
    if (nb + 16 <= n) {
#pragma unroll
      for (int v = 0; v < 8; ++v) outp[(size_t)v * 8] = c[v];
    } else {
#pragma unroll
      for (int v = 0; v < 8; ++v) {
        int orow = nb + (hi << 3) + v;
        if (orow < n) z2[(size_t)orow * 8 + r] = c[v];
      }
    }
  }
}

// -------------------- aggregation init: hagg = bias + dis^2 * hlin (self-loop)
template <int F>
__global__ void k_selfbias(const float* __restrict__ hlin, const float* __restrict__ dis,
                           const float* __restrict__ bias, float* __restrict__ hagg, int n) {
  constexpr int C = F / 4;
  int t = blockIdx.x * blockDim.x + threadIdx.x;
  int node = t / C;
  int q    = t % C;
  if (node >= n) return;
  float w  = dis[node] * dis[node];
  float4 h = ((const float4*)(hlin + (size_t)node * F))[q];
  float4 o;
  o.x = bias[q * 4 + 0] + w * h.x;
  o.y = bias[q * 4 + 1] + w * h.y;
  o.z = bias[q * 4 + 2] + w * h.z;
  o.w = bias[q * 4 + 3] + w * h.w;
  ((float4*)(hagg + (size_t)node * F))[q] = o;
}

// ------------------- edge scatter: hagg[dst] += dis[src]*dis[dst] * hlin[src]
// F/4 lanes per edge: each lane gathers one float4 (b128, consecutive lanes
// share the 64B line) then issues 4 no-return L2 float atomics.
template <int F>
__global__ void k_edge_scatter(const int* __restrict__ src, const int* __restrict__ dst,
                               const float* __restrict__ dis,
                               const float* __restrict__ hlin, float* __restrict__ hagg, int E) {
  constexpr int C = F / 4;
  int t = blockIdx.x * blockDim.x + threadIdx.x;
  int e = t / C;
  int q = t % C;
  if (e >= E) return;
  int s = src[e], d = dst[e];
  float w  = dis[s] * dis[d];
  float4 h = ((const float4*)(hlin + (size_t)s * F))[q];
  float* o = hagg + (size_t)d * F + q * 4;
  atomic_add_f32_noret(o + 0, w * h.x);
  atomic_add_f32_noret(o + 1, w * h.y);
  atomic_add_f32_noret(o + 2, w * h.z);
  atomic_add_f32_noret(o + 3, w * h.w);
}

// ------------------------------------------ final: out = relu(h2agg) @ Wl + bl
__global__ void k_final(const float* __restrict__ h2agg, const float* __restrict__ Wl,
                        const float* __restrict__ bl, float* __restrict__ out, int n) {
  int i = blockIdx.x * blockDim.x + threadIdx.x;
  if (i >= n) return;
  float acc = bl[0];
#pragma unroll
  for (int f = 0; f < 8; ++f)
    acc += fmaxf(h2agg[(size_t)i * 8 + f], 0.0f) * Wl[f];
  out[i] = acc;
}

static inline size_t align_f(size_t v) { return (v + 63) & ~(size_t)63; }

extern "C" void kernel_launch(void* const* d_in, const int* in_sizes, int n_in,
                              void* d_out, int out_size, void* d_ws, size_t ws_size,
                              hipStream_t stream) {
  const float* x  = (const float*)d_in[0];
  const int*   ei = (const int*)d_in[1];
  const float* W1 = (const float*)d_in[2];
  const float* b1 = (const float*)d_in[3];
  const float* W2 = (const float*)d_in[4];
  const float* b2 = (const float*)d_in[5];
  const float* Wl = (const float*)d_in[6];
  const float* bl = (const float*)d_in[7];
  float* out = (float*)d_out;

  const int N = in_sizes[0] / 3;
  const int E = in_sizes[1] / 2;
  const int* src = ei;       // edge_index[0]
  const int* dst = ei + E;   // edge_index[1]

  // workspace carve-up (~20 MB, L2-resident): dis | h1lin | h1agg | z2 | h2agg
  float* ws = (float*)d_ws;
  size_t off = 0;
  float* deg   = ws + off; off += align_f((size_t)N);        // becomes dis in-place
  float* h1lin = ws + off; off += align_f((size_t)N * 16);
  float* h1agg = ws + off; off += align_f((size_t)N * 16);
  float* z2    = ws + off; off += align_f((size_t)N * 8);
  float* h2agg = ws + off; off += align_f((size_t)N * 8);
  (void)ws_size; (void)n_in; (void)out_size;

  const int B = 256;                 // 8 wave32s per block
  const int wpb = B / 32;
  const int ntiles = (N + 15) / 16;  // 6250 exactly for N=100000

  k_init_deg<<<(N + B - 1) / B, B, 0, stream>>>(deg, N);
  k_deg_edges<<<(E + B - 1) / B, B, 0, stream>>>(dst, deg, E);
  k_rsqrt_inplace<<<(N + B - 1) / B, B, 0, stream>>>(deg, N);   // deg -> dis

  k_xw1<<<(ntiles + wpb - 1) / wpb, B, 0, stream>>>(x, W1, h1lin, N, ntiles);
  {
    int T = N * 4;
    k_selfbias<16><<<(T + B - 1) / B, B, 0, stream>>>(h1lin, deg, b1, h1agg, N);
  }
  {
    long long T = (long long)E * 4;
    k_edge_scatter<16><<<(int)((T + B - 1) / B), B, 0, stream>>>(src, dst, deg, h1lin, h1agg, E);
  }

  k_h2<<<(ntiles + wpb - 1) / wpb, B, 0, stream>>>(h1agg, W2, z2, N, ntiles);
  {
    int T = N * 2;
    k_selfbias<8><<<(T + B - 1) / B, B, 0, stream>>>(z2, deg, b2, h2agg, N);
  }
  {
    long long T = (long long)E * 2;
    k_edge_scatter<8><<<(int)((T + B - 1) / B), B, 0, stream>>>(src, dst, deg, z2, h2agg, E);
  }

  k_final<<<(N + B - 1) / B, B, 0, stream>>>(h2agg, Wl, bl, out, N);
}